// CubicSpline_56470230008134
// MI455X (gfx1250) — compile-verified
//
#include <hip/hip_runtime.h>

typedef float v2f __attribute__((ext_vector_type(2)));
typedef float v8f __attribute__((ext_vector_type(8)));

#define D_MODEL 128
#define K_TERMS 63  // NUM_CP - 1

// ---------------------------------------------------------------------------
// Kernel 1: fold the 63 triangular-weighted control points into two D_MODEL
// vectors:  out[n] = Avec + t_n * Bvec.
//   Avec[d] = sum_i (1 - i) * cp[i][d]
//   Bvec[d] = -cp[0][d] + sum_{i>=1} cp[i][d]
// One block of 128 threads; double accumulation (free at this size).
// ---------------------------------------------------------------------------
__global__ void spline_fold_kernel(const float* __restrict__ cp,
                                   float* __restrict__ AB) {
    int d = threadIdx.x;
    if (d >= D_MODEL) return;
    double a = 0.0, b = 0.0;
    for (int i = 0; i < K_TERMS; ++i) {
        double v = (double)cp[i * D_MODEL + d];
        a += (1.0 - (double)i) * v;
        b += (i == 0 ? -1.0 : 1.0) * v;
    }
    AB[d]           = (float)a;
    AB[d + D_MODEL] = (float)b;
}

// ---------------------------------------------------------------------------
// Kernel 2: streaming affine map via V_WMMA_F32_16X16X4_F32.
// Each wave32 produces a 16x128 output slab (16 t-values x all channels)
// as 8 WMMA tiles:
//   A operand (16x4): only K=0 row nonzero = t_m   (VGPR0 lanes 0-15, per ISA)
//   B operand (4x16): Bvec[col] broadcast into every K row (layout-robust,
//                     since A's K=1..3 entries are zero)
//   C accumulator   : Avec[col] in all 8 VGPRs (C/D layout: N = lane&15)
// => D[M,N] = t_M * Bvec[N] + Avec[N].
// D tile stored with non-temporal b32 stores (512 MB stream >> 192 MB L2).
// ---------------------------------------------------------------------------
__global__ void spline_wmma_kernel(const float* __restrict__ t,
                                   const float* __restrict__ AB,
                                   float* __restrict__ out,
                                   int n_tiles) {
    const int lane = threadIdx.x & 31;
    const int wave = threadIdx.x >> 5;
    const int tile = blockIdx.x * (blockDim.x >> 5) + wave;
    if (tile >= n_tiles) return;  // wave-uniform: EXEC stays all-ones for WMMA

    const int  l15  = lane & 15;
    const bool lo   = lane < 16;
    const int  row0 = tile * 16;

    // A operand: 16x4 f32 -> 2 VGPRs.
    //   a.x : K=0 (lanes 0-15) = t_m ; K=2 (lanes 16-31) = 0
    //   a.y : K=1 / K=3 = 0
    float tv = t[row0 + l15];
    tv = fminf(fmaxf(tv, 0.0f), 1.0f);
    v2f a;
    a.x = lo ? tv : 0.0f;
    a.y = 0.0f;

    const float* Avec = AB;
    const float* Bvec = AB + D_MODEL;
    float* orow = out + (size_t)row0 * D_MODEL;
    const int mbase = lo ? 0 : 8;

#pragma unroll
    for (int c = 0; c < 8; ++c) {
        const float av = Avec[c * 16 + l15];
        const float bv = Bvec[c * 16 + l15];

        // B operand: every (K,N) entry in this lane's column = Bvec[N].
        v2f b;
        b.x = bv;
        b.y = bv;

        // C accumulator: Avec[N] replicated across all M rows.
        v8f acc = {av, av, av, av, av, av, av, av};

        // 8 args: (neg_a, A, neg_b, B, c_mod, C, reuse_a, reuse_b)
        acc = __builtin_amdgcn_wmma_f32_16x16x4_f32(
            false, a, false, b, (short)0, acc, false, false);

        // D layout: acc[v] at lane l -> (M = v + (l<16?0:8), N = l&15)
        float* p = orow + (size_t)mbase * D_MODEL + c * 16 + l15;
#pragma unroll
        for (int v = 0; v < 8; ++v) {
            __builtin_nontemporal_store(acc[v], p + (size_t)v * D_MODEL);
        }
    }
}

// ---------------------------------------------------------------------------
// Kernel 3: scalar tail for rows not covered by full 16-row tiles
// (N = 1,000,000 is divisible by 16, so normally never launched).
// ---------------------------------------------------------------------------
__global__ void spline_tail_kernel(const float* __restrict__ t,
                                   const float* __restrict__ AB,
                                   float* __restrict__ out,
                                   int start_row, int n_rows) {
    int idx = blockIdx.x * blockDim.x + threadIdx.x;
    int row = start_row + idx / D_MODEL;
    int d   = idx % D_MODEL;
    if (row >= n_rows) return;
    float tv = fminf(fmaxf(t[row], 0.0f), 1.0f);
    out[(size_t)row * D_MODEL + d] = fmaf(tv, AB[D_MODEL + d], AB[d]);
}

extern "C" void kernel_launch(void* const* d_in, const int* in_sizes, int n_in,
                              void* d_out, int out_size, void* d_ws, size_t ws_size,
                              hipStream_t stream) {
    const float* t  = (const float*)d_in[0];   // [N] fp32
    const float* cp = (const float*)d_in[1];   // [64, 128] fp32
    float*       out = (float*)d_out;          // [N, 128] fp32
    float*       AB  = (float*)d_ws;           // 256 floats scratch
    const int N = in_sizes[0];

    // Fold control points -> Avec/Bvec (one tiny block; ordered on stream).
    spline_fold_kernel<<<1, D_MODEL, 0, stream>>>(cp, AB);

    // Main streaming WMMA kernel: 8 waves per 256-thread block, 1 tile/wave.
    const int n_tiles = N / 16;
    if (n_tiles > 0) {
        const int waves_per_block = 256 / 32;
        const int grid = (n_tiles + waves_per_block - 1) / waves_per_block;
        spline_wmma_kernel<<<grid, 256, 0, stream>>>(t, AB, out, n_tiles);
    }

    // Tail rows (none for N = 1e6).
    const int rem = N - n_tiles * 16;
    if (rem > 0) {
        const int threads = rem * D_MODEL;
        spline_tail_kernel<<<(threads + 255) / 256, 256, 0, stream>>>(
            t, AB, out, n_tiles * 16, N);
    }
}